// Model_13984413516353
// MI455X (gfx1250) — compile-verified
//
#include <hip/hip_runtime.h>
#include <math.h>

#define BT 4
#define NN 50000
#define MM 256
#define FF 256
#define GAMMA_C 4.2f
#define SLOPE_C 0.2f

// padded row stride (in float2 units) for the interleaved W band in LDS:
// half-wave (kk+2) offset = 144 float2 = 288 dwords == 32 mod 64 banks -> conflict-free
#define WB_COLS 128
#define WB_STRIDE 144

typedef __attribute__((ext_vector_type(2))) float v2f;
typedef __attribute__((ext_vector_type(8))) float v8f;

// ---------- helpers ----------

__device__ __forceinline__ void atomicMaxF(float* addr, float val) {
    // works with init value -inf: positive floats via signed-int max,
    // negative floats via unsigned-int min (bit patterns are order-reversed)
    if (val >= 0.0f) atomicMax((int*)addr, __float_as_int(val));
    else             atomicMin((unsigned int*)addr, __float_as_uint(val));
}

__device__ __forceinline__ float block_reduce_sum(float v, float* sh) {
    int t = threadIdx.x;
    sh[t] = v;
    __syncthreads();
    for (int s = blockDim.x >> 1; s > 0; s >>= 1) {
        if (t < s) sh[t] += sh[t + s];
        __syncthreads();
    }
    float r = sh[0];
    __syncthreads();
    return r;
}

// ---------- kernel 0: init ----------
__global__ void k_init(float* out, long long outN, float* es, float* oe,
                       float* nmax, float* nsum, float* Dd, float* Sm, float* sc) {
    long long i = (long long)blockIdx.x * blockDim.x + threadIdx.x;
    long long stride = (long long)gridDim.x * blockDim.x;
    for (long long j = i; j < outN; j += stride) out[j] = 0.0f;
    for (long long j = i; j < (long long)MM * BT * FF; j += stride) { es[j] = 0.0f; oe[j] = 0.0f; }
    for (long long j = i; j < (long long)NN * BT; j += stride) { nmax[j] = -__builtin_inff(); nsum[j] = 0.0f; }
    for (long long j = i; j < NN; j += stride) Dd[j] = 0.0f;
    for (long long j = i; j < MM; j += stride) Sm[j] = 0.0f;
    if (i < 2) sc[i] = 0.0f;
}

// ---------- kernel 1: x1[n,b,f] = sum_k x[b,n,k] * W[k,f]  (f32 WMMA) ----------
// grid: (ceil(3125/4), 2 f-bands of 128, BT), block = 128 (4 waves, one 16-row tile each)
// W band staged in LDS k-pair-interleaved + padded: one ds_load_b64 per WMMA B fragment.
__global__ void k_gemm(const float* __restrict__ X, const float* __restrict__ W,
                       float* __restrict__ x1) {
    __shared__ v2f wl2[(FF / 2) * WB_STRIDE];        // 128 k-pairs x 144 float2 = 144 KB
    int b  = blockIdx.z;
    int fb = blockIdx.y * WB_COLS;

    // stage W[k, fb+f] -> wl2[(k/2)*WB_STRIDE + f][k&1]
    float* wlf = (float*)wl2;
    for (int idx = threadIdx.x; idx < FF * WB_COLS; idx += blockDim.x) {
        int k = idx >> 7, f = idx & (WB_COLS - 1);
        wlf[((k >> 1) * WB_STRIDE + f) * 2 + (k & 1)] = W[k * FF + fb + f];
    }
    __syncthreads();

    int wave = threadIdx.x >> 5;
    int lane = threadIdx.x & 31;
    int tile = blockIdx.x * 4 + wave;
    int n0 = tile * 16;
    if (n0 >= NN) return;                   // whole wave drops out; EXEC stays all-ones for WMMA
    int half = lane >> 4;                   // 0: lanes 0-15, 1: lanes 16-31
    int l16  = lane & 15;

    const float* Xrow = X + ((long long)b * NN + (n0 + l16)) * FF;

    v8f acc[8];
    #pragma unroll
    for (int ft = 0; ft < 8; ++ft)
        #pragma unroll
        for (int v = 0; v < 8; ++v) acc[ft][v] = 0.0f;

    for (int k = 0; k < FF; k += 4) {
        int p = (k >> 1) + half;            // k-pair index for this half-wave
        v2f a = *(const v2f*)(Xrow + k + 2 * half);  // A[M=l16, K=kk], A[M=l16, K=kk+1]
        #pragma unroll
        for (int ft = 0; ft < 8; ++ft) {
            v2f bb = wl2[p * WB_STRIDE + ft * 16 + l16];  // (B[K=kk,col], B[K=kk+1,col])
            acc[ft] = __builtin_amdgcn_wmma_f32_16x16x4_f32(
                false, a, false, bb, (short)0, acc[ft], false, false);
        }
    }
    #pragma unroll
    for (int ft = 0; ft < 8; ++ft) {
        int col = fb + ft * 16 + l16;
        #pragma unroll
        for (int v = 0; v < 8; ++v) {
            int r = n0 + v + 8 * half;      // C/D layout: lanes 16-31 hold M=v+8
            x1[(long long)r * (BT * FF) + b * FF + col] = acc[ft][v];
        }
    }
}

// ---------- kernel 2: edge_sums[m,b,f] (edge entries are contiguous) ----------
__global__ void k_edge_sum(const float* __restrict__ x1, const int* __restrict__ node_idx,
                           float* __restrict__ es, int Le) {
    int m = blockIdx.x, b = blockIdx.y, f = threadIdx.x;
    int base = m * Le;
    float acc = 0.0f;
    for (int i = 0; i < Le; ++i) {
        int n = node_idx[base + i];
        acc += x1[(long long)n * (BT * FF) + b * FF + f];
    }
    es[(m * BT + b) * FF + f] = acc;
}

// ---------- kernel 3: per-node att-dot t[n,b] and row-sum R[n] ----------
__global__ void k_node_scalar(const float* __restrict__ x1, const float* __restrict__ att,
                              float* __restrict__ t, float* __restrict__ R) {
    __shared__ float sh[256];
    int n = blockIdx.x, f = threadIdx.x;
    const float* xr = x1 + (long long)n * (BT * FF);
    float tb[BT];
    float r = 0.0f;
    #pragma unroll
    for (int b = 0; b < BT; ++b) {
        float v = xr[b * FF + f];
        tb[b] = v * att[f];
        r += v;
    }
    #pragma unroll
    for (int b = 0; b < BT; ++b) {
        float s = block_reduce_sum(tb[b], sh);
        if (f == 0) t[n * BT + b] = s;
    }
    float rs = block_reduce_sum(r, sh);
    if (f == 0) R[n] = rs;
}

// ---------- kernel 4: per-edge att-dot s[m,b], norms, total sums S[m] ----------
__global__ void k_edge_scalar(const float* __restrict__ es, const float* __restrict__ att,
                              float* __restrict__ s, float* __restrict__ Sm,
                              float* __restrict__ nrm) {
    __shared__ float sh[256];
    int m = blockIdx.x, b = blockIdx.y, f = threadIdx.x;
    float v = es[(m * BT + b) * FF + f];
    float sd = block_reduce_sum(v * att[FF + f], sh);
    float sq = block_reduce_sum(v * v, sh);
    float ss = block_reduce_sum(v, sh);
    if (f == 0) {
        s[m * BT + b] = sd;
        nrm[m * BT + b] = sqrtf(sq);
        atomicAdd(&Sm[m], ss);
    }
}

// ---------- kernel 5a: logits + leaky_relu + node max + degree + constrain-diff ----------
__global__ void k_logits(const int* __restrict__ node_idx, const int* __restrict__ edge_idx,
                         const float* __restrict__ t, const float* __restrict__ s,
                         const float* __restrict__ R, const float* __restrict__ Sm,
                         float* __restrict__ abuf, float* __restrict__ nmax,
                         float* __restrict__ Dd, float* __restrict__ sc, int E) {
    __shared__ float sh[256];
    int tid = blockIdx.x * blockDim.x + threadIdx.x;
    int stride = gridDim.x * blockDim.x;
    float diff = 0.0f;
    for (int e = tid; e < E; e += stride) {
        int n = node_idx[e], m = edge_idx[e];
        atomicAdd(&Dd[n], 1.0f);
        diff += R[n] - Sm[m];
        #pragma unroll
        for (int b = 0; b < BT; ++b) {
            float a = t[n * BT + b] + s[m * BT + b];
            a = (a >= 0.0f) ? a : SLOPE_C * a;
            abuf[(long long)e * BT + b] = a;
            atomicMaxF(&nmax[n * BT + b], a);
        }
    }
    float d = block_reduce_sum(diff, sh);
    if (threadIdx.x == 0) atomicAdd(&sc[0], d);
}

// ---------- kernel 5b: exp + node sum ----------
__global__ void k_expsum(const int* __restrict__ node_idx, float* __restrict__ abuf,
                         const float* __restrict__ nmax, float* __restrict__ nsum, int E) {
    int tid = blockIdx.x * blockDim.x + threadIdx.x;
    int stride = gridDim.x * blockDim.x;
    for (int e = tid; e < E; e += stride) {
        int n = node_idx[e];
        #pragma unroll
        for (int b = 0; b < BT; ++b) {
            float ev = expf(abuf[(long long)e * BT + b] - nmax[n * BT + b]);
            abuf[(long long)e * BT + b] = ev;
            atomicAdd(&nsum[n * BT + b], ev);
        }
    }
}

// ---------- kernel 6: out_e[m,b,f] = (1/Le) * sum alpha * x1[node]  ----------
__global__ void k_out_e(const float* __restrict__ x1, const int* __restrict__ node_idx,
                        const float* __restrict__ abuf, const float* __restrict__ nsum,
                        float* __restrict__ oe, int Le) {
    __shared__ float alph[1024];
    __shared__ int   nodes[1024];
    int m = blockIdx.x, b = blockIdx.y, f = threadIdx.x;
    int base = m * Le;
    for (int i = f; i < Le; i += blockDim.x) {
        int n = node_idx[base + i];
        nodes[i] = n;
        alph[i] = abuf[(long long)(base + i) * BT + b] / nsum[n * BT + b];
    }
    __syncthreads();
    float acc = 0.0f;
    for (int i = 0; i < Le; ++i)
        acc += alph[i] * x1[(long long)nodes[i] * (BT * FF) + b * FF + f];
    oe[(m * BT + b) * FF + f] = acc / (float)Le;
}

// ---------- kernel 7: scatter to nodes, directly into out[b,n,f] ----------
__global__ void k_scatter(const int* __restrict__ node_idx, const int* __restrict__ edge_idx,
                          const float* __restrict__ abuf, const float* __restrict__ nsum,
                          const float* __restrict__ Dd, const float* __restrict__ oe,
                          float* __restrict__ out, int E) {
    int e = blockIdx.x;
    int f = threadIdx.x;
    int n = node_idx[e], m = edge_idx[e];
    float dN = Dd[n];
    #pragma unroll
    for (int b = 0; b < BT; ++b) {
        float coef = dN * abuf[(long long)e * BT + b] / nsum[n * BT + b];
        atomicAdd(&out[((long long)b * NN + n) * FF + f], coef * oe[(m * BT + b) * FF + f]);
    }
}

// ---------- kernel 8: O(M^2) pairwise hyperedge loss via f32 WMMA Gram matrix ----------
// grid (16,16), one wave (32 threads) per 16x16 tile of the [M,M] Gram matrix
__global__ void k_loss(const float* __restrict__ es, const float* __restrict__ nrm,
                       float* __restrict__ sc) {
    int k0 = blockIdx.x * 16, m0 = blockIdx.y * 16;
    int lane = threadIdx.x & 31;
    int half = lane >> 4, l16 = lane & 15;

    float li[8];
    #pragma unroll
    for (int v = 0; v < 8; ++v) li[v] = 0.0f;

    for (int b = 0; b < BT; ++b) {
        v8f g;
        #pragma unroll
        for (int v = 0; v < 8; ++v) g[v] = 0.0f;
        const float* Ak = es + ((long long)(k0 + l16) * BT + b) * FF;
        const float* Am = es + ((long long)(m0 + l16) * BT + b) * FF;
        for (int fi = 0; fi < FF; fi += 4) {
            int kk = fi + 2 * half;
            v2f a  = *(const v2f*)(Ak + kk);   // A[M=k-row, K]
            v2f bb = *(const v2f*)(Am + kk);   // B[K, N=m-col] (row of es == column of es^T)
            g = __builtin_amdgcn_wmma_f32_16x16x4_f32(
                false, a, false, bb, (short)0, g, false, false);
        }
        #pragma unroll
        for (int v = 0; v < 8; ++v) {
            int kr = k0 + v + 8 * half;
            int mc = m0 + l16;
            float nk = nrm[kr * BT + b], nm = nrm[mc * BT + b];
            float G = g[v];
            float cosv = G / (nk * nm);
            float d2 = nk * nk + nm * nm - 2.0f * G;
            float dist = sqrtf(fmaxf(d2, 1e-12f));
            li[v] += cosv * dist + (1.0f - cosv) * fmaxf(GAMMA_C - dist, 0.0f);
        }
    }
    float sum = 0.0f;
    #pragma unroll
    for (int v = 0; v < 8; ++v) sum += fabsf(li[v] * (1.0f / BT));  // |mean over b|
    for (int off = 16; off > 0; off >>= 1) sum += __shfl_down(sum, off);
    if (lane == 0) atomicAdd(&sc[1], sum);
}

// ---------- kernel 9: finalize constrain scalar ----------
__global__ void k_final(const float* __restrict__ sc, float* __restrict__ out, int E) {
    float inv = 1.0f / ((float)E * (float)BT * (float)FF);
    out[(long long)BT * NN * FF] =
        fabsf(sc[0] * inv) + sc[1] / ((float)(MM + 1) * (float)(MM + 1));
}

// ---------- launch ----------
extern "C" void kernel_launch(void* const* d_in, const int* in_sizes, int n_in,
                              void* d_out, int out_size, void* d_ws, size_t ws_size,
                              hipStream_t stream) {
    const float* X        = (const float*)d_in[0];   // [BT, N, F]
    const float* W        = (const float*)d_in[1];   // [F, F]
    const float* att      = (const float*)d_in[2];   // [2F]
    const int*   node_idx = (const int*)d_in[3];     // [E]
    const int*   edge_idx = (const int*)d_in[4];     // [Ee]
    float* out = (float*)d_out;

    int En = in_sizes[3], Ee = in_sizes[4];
    int E = En < Ee ? En : Ee;
    int Le = E / MM;
    E = Le * MM;                      // entries fully covered by hyperedges

    float* ws = (float*)d_ws;
    long long NX  = (long long)NN * BT * FF;
    long long MBF = (long long)MM * BT * FF;
    long long NBT = (long long)NN * BT;
    long long MBT = (long long)MM * BT;
    long long EBT = (long long)E * BT;

    float* x1   = ws;                  long long o = NX;
    float* es   = ws + o;              o += MBF;
    float* oe   = ws + o;              o += MBF;
    float* t    = ws + o;              o += NBT;
    float* R    = ws + o;              o += NN;
    float* s    = ws + o;              o += MBT;
    float* Sm   = ws + o;              o += MM;
    float* nrm  = ws + o;              o += MBT;
    float* abuf = ws + o;              o += EBT;
    float* nmax = ws + o;              o += NBT;
    float* nsum = ws + o;              o += NBT;
    float* Dd   = ws + o;              o += NN;
    float* sc   = ws + o;              o += 2;

    long long outN = NX + 1;

    k_init<<<8192, 256, 0, stream>>>(out, outN, es, oe, nmax, nsum, Dd, Sm, sc);

    dim3 gg((3125 + 3) / 4, 2, BT);
    k_gemm<<<gg, 128, 0, stream>>>(X, W, x1);

    k_node_scalar<<<NN, 256, 0, stream>>>(x1, att, t, R);
    k_edge_sum<<<dim3(MM, BT), 256, 0, stream>>>(x1, node_idx, es, Le);
    k_edge_scalar<<<dim3(MM, BT), 256, 0, stream>>>(es, att, s, Sm, nrm);

    k_logits<<<1024, 256, 0, stream>>>(node_idx, edge_idx, t, s, R, Sm, abuf, nmax, Dd, sc, E);
    k_expsum<<<1024, 256, 0, stream>>>(node_idx, abuf, nmax, nsum, E);

    k_out_e<<<dim3(MM, BT), 256, 0, stream>>>(x1, node_idx, abuf, nsum, oe, Le);
    k_scatter<<<E, 256, 0, stream>>>(node_idx, edge_idx, abuf, nsum, Dd, oe, out, E);

    k_loss<<<dim3(16, 16), 32, 0, stream>>>(es, nrm, sc);
    k_final<<<1, 1, 0, stream>>>(sc, out, E);
}